// AdjacencyAwareMultiHeadAttention_34454227648598
// MI455X (gfx1250) — compile-verified
//
#include <hip/hip_runtime.h>
#include <hip/hip_bf16.h>

// Problem constants (B, N, IN_DIM, D, H) = (8, 1024, 128, 32, 8)
#define BB_    8
#define NN_    1024
#define IND_   128
#define DD_    32
#define HH_    8
#define HD_    256           // H*D
#define QTILES (NN_/16)      // 64
#define NWG_ATT (BB_*HH_*QTILES)   // 4096
#define PITCH  1028          // LDS row pitch (floats): conflict-free b64 A-operand reads

typedef __attribute__((ext_vector_type(2))) float v2f;
typedef __attribute__((ext_vector_type(8))) float v8f;

static __device__ __forceinline__ v8f wmma_f32(v2f a, v2f b, v8f c) {
  // V_WMMA_F32_16X16X4_F32 : D(16x16 f32) = A(16x4 f32) * B(4x16 f32) + C
  return __builtin_amdgcn_wmma_f32_16x16x4_f32(
      /*neg_a=*/false, a, /*neg_b=*/false, b,
      /*c_mod=*/(short)0, c, /*reuse_a=*/false, /*reuse_b=*/false);
}

// ---------------------------------------------------------------- zero counts
__global__ void aamha_zero_kernel(unsigned* counts) {
  if (threadIdx.x < 2) counts[threadIdx.x] = 0u;
}

// ------------------------------------------------------- Q/K/V projection GEMM
// [8192,128] x [128,256] + bias, 3 matrices, one 16x16 tile per wave.
// Outputs (per (b,h) block):
//   Q  : [n][d]                       (row-major, A-operand friendly)
//   Kt2: [(d>>1)][n][d&1]             (pair-interleaved -> float2 B-operand)
//   V2 : [(n>>1)][d][n&1]             (pair-interleaved -> float2 B-operand)
__global__ void aamha_proj_kernel(const float* __restrict__ h,
                                  const float* __restrict__ Wq, const float* __restrict__ bq,
                                  const float* __restrict__ Wk, const float* __restrict__ bk,
                                  const float* __restrict__ Wv, const float* __restrict__ bv,
                                  float* __restrict__ Qo, float* __restrict__ Kto,
                                  float* __restrict__ Vo) {
  const int tid  = threadIdx.x;
  const int lane = tid & 31;
  const int wave = tid >> 5;
  const int tile = blockIdx.x * 2 + wave;          // 0 .. 24575
  const int mat  = tile / 8192;                    // 0=Q 1=K 2=V
  const int t    = tile - mat * 8192;
  const int m0   = (t >> 4) * 16;                  // row tile (over B*N)
  const int n0   = (t & 15) * 16;                  // col tile (over H*D)

  const float* W    = (mat == 0) ? Wq : (mat == 1) ? Wk : Wv;
  const float* bias = (mat == 0) ? bq : (mat == 1) ? bk : bv;

  const int l15  = lane & 15;
  const int koff = (lane < 16) ? 0 : 2;
  const float* arow = h + (size_t)(m0 + l15) * IND_;
  const float* brow = W + (size_t)(n0 + l15) * IND_;   // B[k][n] = W[n][k]

  v8f c = {};
#pragma unroll
  for (int k0 = 0; k0 < IND_; k0 += 4) {
    v2f a = *(const v2f*)(arow + k0 + koff);
    v2f b = *(const v2f*)(brow + k0 + koff);
    c = wmma_f32(a, b, c);
  }

  const int col = n0 + l15;             // output feature 0..255
  const float bval = bias[col];
  const int hh = col >> 5;              // head
  const int d  = col & 31;
  const int mbase = (lane >= 16) ? 8 : 0;
#pragma unroll
  for (int i = 0; i < 8; ++i) {
    const int g    = m0 + i + mbase;    // global row over B*N
    const int bidx = g >> 10;
    const int n    = g & 1023;
    const float val = c[i] + bval;
    const size_t base = (size_t)(bidx * HH_ + hh) * (NN_ * DD_);
    if (mat == 0) {
      Qo[base + (size_t)n * DD_ + d] = val;
    } else if (mat == 1) {
      Kto[base + (((size_t)(d >> 1) * NN_ + n) << 1) + (d & 1)] = val;
    } else {
      Vo[base + (((size_t)(n >> 1) * DD_ + d) << 1) + (n & 1)] = val;
    }
  }
}

// ------------------------------------------------------- mask counts over A
__global__ void aamha_astats_kernel(const int* __restrict__ A, unsigned* counts) {
  __shared__ unsigned s[2];
  const int tid = threadIdx.x;
  if (tid == 0) { s[0] = 0u; s[1] = 0u; }
  __syncthreads();
  unsigned c1 = 0, c2 = 0;
  const int total = BB_ * NN_ * NN_;
  for (int i = blockIdx.x * blockDim.x + tid; i < total; i += gridDim.x * blockDim.x) {
    const int v = A[i];
    c1 += (v == 1) ? 1u : 0u;
    c2 += (v > 1)  ? 1u : 0u;
  }
  atomicAdd(&s[0], c1);
  atomicAdd(&s[1], c2);
  __syncthreads();
  if (tid == 0) { atomicAdd(&counts[0], s[0]); atomicAdd(&counts[1], s[1]); }
}

// ------------------------------------------------------- fused attention
// One WG (128 threads = 4 waves) per (b, h, 16-query tile).
// Phase A: S = Q K^T via WMMA into LDS (16 x 1024, pitch 1028), key-pad -> -inf.
//          4 waves x 16 key-tiles.
// Phase B: softmax over keys (8 threads per query row).
// Phase C: normalize, zero padded query rows, masked sums (pre), *alpha^A,
//          masked sums (post), write reweighted probs back to LDS.
// Phase D: out = att @ V. Wave pair (2w,2w+1) computes d-tile w; each wave
//          accumulates half the K range; deterministic LDS combine.
__global__ void aamha_attn_kernel(const float* __restrict__ Q,
                                  const float* __restrict__ Kt,
                                  const float* __restrict__ V,
                                  const int*   __restrict__ A,
                                  const int*   __restrict__ lengths,
                                  const float* __restrict__ alphaPtr,
                                  float* __restrict__ out,
                                  float* __restrict__ partials) {
  extern __shared__ float smem[];
  float* satt = smem;                 // 16 * PITCH
  float* sred = satt + 16 * PITCH;    // 128
  float* bred = sred + 128;           // 512 (block reduce; reused as phase-D exchange)

  const int qt  = blockIdx.x;         // 0..63
  const int hh  = blockIdx.y;         // 0..7
  const int bb  = blockIdx.z;         // 0..7
  const int tid = threadIdx.x;        // 0..127
  const int lane = tid & 31;
  const int wave = tid >> 5;          // 0..3
  const int q0  = qt * 16;
  const int len = lengths[bb];
  const float alpha = alphaPtr[0];

  const size_t bh = (size_t)(bb * HH_ + hh) * (NN_ * DD_);
  const float* Qp  = Q  + bh;
  const float* Ktp = Kt + bh;   // pair-interleaved [(d>>1)][n][2]
  const float* Vp  = V  + bh;   // pair-interleaved [(n>>1)][d][2]

  const int l15  = lane & 15;
  const int koff = (lane < 16) ? 0 : 2;
  const int mbase = (lane >= 16) ? 8 : 0;
  const float NEG_INF = -__builtin_inff();
  const float rscale = 0.17677669529663687f;   // 1/sqrt(D=32)

  // ---- Phase A: S = Q K^T -------------------------------------------------
  v2f aq[8];
  {
    const float* qrow = Qp + (size_t)(q0 + l15) * DD_;
#pragma unroll
    for (int j = 0; j < 8; ++j) aq[j] = *(const v2f*)(qrow + 4 * j + koff);
  }
  for (int kt2 = wave * 16; kt2 < wave * 16 + 16; ++kt2) {
    const int key0 = kt2 * 16;
    v8f c = {};
#pragma unroll
    for (int j = 0; j < 8; ++j) {
      const int dd = 4 * j + koff;                     // even
      v2f b = *(const v2f*)(Ktp + (((size_t)(dd >> 1) * NN_ + key0 + l15) << 1));
      c = wmma_f32(aq[j], b, c);
    }
    const int col = key0 + l15;
    const bool kpad = (col >= len);
#pragma unroll
    for (int i = 0; i < 8; ++i) {
      float sv = c[i] * rscale;
      if (kpad) sv = NEG_INF;
      satt[(i + mbase) * PITCH + col] = sv;
    }
  }
  __syncthreads();

  // ---- Phase B: softmax ---------------------------------------------------
  const int r   = tid & 15;           // query row within tile
  const int seg = tid >> 4;           // 8 segments of 128 keys
  const int c0  = seg * 128;
  float m = NEG_INF;
  for (int cc = 0; cc < 128; ++cc) m = fmaxf(m, satt[r * PITCH + c0 + cc]);
  sred[r * 8 + seg] = m;
  __syncthreads();
  float rmax = sred[r * 8];
#pragma unroll
  for (int i = 1; i < 8; ++i) rmax = fmaxf(rmax, sred[r * 8 + i]);
  float ssum = 0.f;
  for (int cc = 0; cc < 128; ++cc) {
    const int idx = r * PITCH + c0 + cc;
    const float e = __expf(satt[idx] - rmax);
    satt[idx] = e;
    ssum += e;
  }
  __syncthreads();
  sred[r * 8 + seg] = ssum;
  __syncthreads();
  float rowsum = sred[r * 8];
#pragma unroll
  for (int i = 1; i < 8; ++i) rowsum += sred[r * 8 + i];
  const float rinv = 1.0f / rowsum;

  // ---- Phase C: normalize, pad-zero, masked sums, alpha^A reweight --------
  const bool qpad = (q0 + r) >= len;
  const float lg2a = log2f(alpha);
  const int* Arow = A + ((size_t)(bb * NN_ + q0 + r)) * NN_ + c0;
  float p1 = 0.f, p2 = 0.f, w1 = 0.f, w2 = 0.f;
  for (int cc = 0; cc < 128; ++cc) {
    const int idx = r * PITCH + c0 + cc;
    float p = satt[idx] * rinv;
    p = qpad ? 0.f : p;
    const int av = Arow[cc];
    const float w = exp2f((float)av * lg2a);   // alpha^A  (exp(A log a), as jnp.power)
    const float pw = p * w;
    const bool is1 = (av == 1), isg = (av > 1);
    p1 += is1 ? p  : 0.f;
    p2 += isg ? p  : 0.f;
    w1 += is1 ? pw : 0.f;
    w2 += isg ? pw : 0.f;
    satt[idx] = pw;
  }
  bred[tid]        = p1;
  bred[128 + tid]  = p2;
  bred[256 + tid]  = w1;
  bred[384 + tid]  = w2;
  __syncthreads();
  if (tid == 0) {
    float s1 = 0.f, s2 = 0.f, s3 = 0.f, s4 = 0.f;
    for (int i = 0; i < 128; ++i) {
      s1 += bred[i]; s2 += bred[128 + i]; s3 += bred[256 + i]; s4 += bred[384 + i];
    }
    const int wg = (bb * HH_ + hh) * QTILES + qt;
    partials[wg * 4 + 0] = s1;
    partials[wg * 4 + 1] = s2;
    partials[wg * 4 + 2] = s3;
    partials[wg * 4 + 3] = s4;
  }
  __syncthreads();

  // ---- Phase D: out = att @ V --------------------------------------------
  const int dtile = wave >> 1;        // 0 or 1 -> d columns [16*dtile, 16*dtile+16)
  const int khalf = wave & 1;         // key range half
  const int n0 = dtile * 16;
  const int d  = n0 + l15;
  v8f c = {};
  const int kbeg = khalf * 512, kend = kbeg + 512;
  for (int k0 = kbeg; k0 < kend; k0 += 4) {
    const int kk = k0 + koff;                          // even
    v2f a = *(const v2f*)(&satt[l15 * PITCH + kk]);
    v2f b = *(const v2f*)(Vp + (((size_t)(kk >> 1) * DD_ + d) << 1));
    c = wmma_f32(a, b, c);
  }
  // deterministic cross-wave combine: khalf==1 publishes, khalf==0 adds.
  if (khalf == 1) {
#pragma unroll
    for (int i = 0; i < 8; ++i) bred[dtile * 256 + lane * 8 + i] = c[i];
  }
  __syncthreads();
  if (khalf == 0) {
    float* op = out + bh;
#pragma unroll
    for (int i = 0; i < 8; ++i) {
      const float v = c[i] + bred[dtile * 256 + lane * 8 + i];
      op[(size_t)(q0 + i + mbase) * DD_ + d] = v;
    }
  }
}

// ------------------------------------------------------- final scalar reduce
__global__ void aamha_finalize_kernel(const float* __restrict__ partials,
                                      const unsigned* __restrict__ counts,
                                      float* __restrict__ out_tail) {
  if (threadIdx.x == 0 && blockIdx.x == 0) {
    float s1 = 0.f, s2 = 0.f, s3 = 0.f, s4 = 0.f;
    for (int i = 0; i < NWG_ATT; ++i) {       // fixed order -> deterministic
      s1 += partials[4 * i + 0];
      s2 += partials[4 * i + 1];
      s3 += partials[4 * i + 2];
      s4 += partials[4 * i + 3];
    }
    const float c1 = (float)HH_ * (float)counts[0];
    const float c2 = (float)HH_ * (float)counts[1];
    out_tail[0] = s1 / c1;   // pre_d1
    out_tail[1] = s2 / c2;   // pre_d2
    out_tail[2] = s3 / c1;   // post_d1
    out_tail[3] = s4 / c2;   // post_d2
  }
}

extern "C" void kernel_launch(void* const* d_in, const int* in_sizes, int n_in,
                              void* d_out, int out_size, void* d_ws, size_t ws_size,
                              hipStream_t stream) {
  (void)in_sizes; (void)n_in; (void)out_size; (void)ws_size;
  const float* h       = (const float*)d_in[0];
  const int*   A       = (const int*)  d_in[1];
  const int*   lengths = (const int*)  d_in[2];
  const float* alpha   = (const float*)d_in[3];
  const float* Wq      = (const float*)d_in[4];
  const float* bq      = (const float*)d_in[5];
  const float* Wk      = (const float*)d_in[6];
  const float* bk      = (const float*)d_in[7];
  const float* Wv      = (const float*)d_in[8];
  const float* bv      = (const float*)d_in[9];
  float* out = (float*)d_out;

  // workspace layout
  const size_t QSZ = (size_t)BB_ * HH_ * NN_ * DD_;   // 2,097,152 floats each
  float* wsf  = (float*)d_ws;
  float* Qb   = wsf;
  float* Ktb  = wsf + QSZ;
  float* Vb   = wsf + 2 * QSZ;
  float* part = wsf + 3 * QSZ;                        // 4096*4 floats
  unsigned* counts = (unsigned*)(part + (size_t)NWG_ATT * 4);

  aamha_zero_kernel<<<1, 32, 0, stream>>>(counts);

  // 3 * 512 * 16 tiles, 2 waves per WG
  aamha_proj_kernel<<<12288, 64, 0, stream>>>(h, Wq, bq, Wk, bk, Wv, bv, Qb, Ktb, Vb);

  aamha_astats_kernel<<<512, 256, 0, stream>>>(A, counts);

  const size_t shmem = (size_t)(16 * PITCH + 128 + 512) * sizeof(float);
  aamha_attn_kernel<<<dim3(QTILES, HH_, BB_), 128, shmem, stream>>>(
      Qb, Ktb, Vb, A, lengths, alpha, out, part);

  aamha_finalize_kernel<<<1, 32, 0, stream>>>(part, counts, out + QSZ);
}